// HPSS_86191403696249
// MI455X (gfx1250) — compile-verified
//
#include <hip/hip_runtime.h>
#include <stdint.h>

// Problem constants (fixed by the reference setup)
constexpr int B_DIM  = 2;
constexpr int C_DIM  = 1;
constexpr int F_DIM  = 1025;
constexpr int T_DIM  = 2048;
constexpr int KWIN   = 31;
constexpr int PAD    = 15;                 // (K-1)/2
constexpr int N_ELEM = B_DIM * C_DIM * F_DIM * T_DIM;   // 4,198,400
constexpr float TINY_F = 1.1754943508222875e-38f;       // np.finfo(f32).tiny

// ---------------------------------------------------------------------------
// Forgetful-selection median of 31 (zero-padded window supplied by `get`).
// 17-register working set; each pass puts min at r[0], max at r[S-1],
// then drops both (max implicitly by shrinking, min by overwrite with the
// next window element). Ends with median-of-3. ~250 compare-exchanges.
// ---------------------------------------------------------------------------
__device__ __forceinline__ void ce(float& a, float& b) {
  float lo = fminf(a, b);
  b = fmaxf(a, b);
  a = lo;
}

template <int S>
__device__ __forceinline__ void minmax_pass(float (&r)[17]) {
#pragma unroll
  for (int i = 1; i < S; ++i) ce(r[0], r[i]);        // min -> r[0]
#pragma unroll
  for (int i = 1; i < S - 1; ++i) ce(r[i], r[S - 1]); // max -> r[S-1]
}

template <typename Get>
__device__ __forceinline__ float median31(const Get& get) {
  float r[17];
#pragma unroll
  for (int i = 0; i < 17; ++i) r[i] = get(i);
  minmax_pass<17>(r); r[0] = get(17);
  minmax_pass<16>(r); r[0] = get(18);
  minmax_pass<15>(r); r[0] = get(19);
  minmax_pass<14>(r); r[0] = get(20);
  minmax_pass<13>(r); r[0] = get(21);
  minmax_pass<12>(r); r[0] = get(22);
  minmax_pass<11>(r); r[0] = get(23);
  minmax_pass<10>(r); r[0] = get(24);
  minmax_pass< 9>(r); r[0] = get(25);
  minmax_pass< 8>(r); r[0] = get(26);
  minmax_pass< 7>(r); r[0] = get(27);
  minmax_pass< 6>(r); r[0] = get(28);
  minmax_pass< 5>(r); r[0] = get(29);
  minmax_pass< 4>(r); r[0] = get(30);
  // live set: r[0..2]
  float lo = fminf(r[0], r[1]);
  float hi = fmaxf(r[0], r[1]);
  return fmaxf(lo, fminf(hi, r[2]));
}

// ---------------------------------------------------------------------------
// Kernel 1: median along T (contiguous axis). One block per (b,c,f) row.
// Row is staged into LDS with gfx1250 async global->LDS copies (ASYNCcnt),
// pads are zeroed with plain DS stores, then 8 medians per thread from LDS.
// ---------------------------------------------------------------------------
__global__ __launch_bounds__(256) void hpss_med_time(const float* __restrict__ S,
                                                     float* __restrict__ harm) {
  __shared__ float tile[T_DIM + 2 * PAD];   // 2078 floats, 8.3 KB
  const unsigned row  = blockIdx.x;         // 0 .. B*C*F-1
  const unsigned tid  = threadIdx.x;
  const size_t rbase  = (size_t)row * T_DIM;

  // zero padding (left and right halo)
  if (tid < PAD) {
    tile[tid] = 0.0f;
    tile[T_DIM + PAD + tid] = 0.0f;
  }

  // async-copy the 2048 interior elements: 8 iterations x 256 lanes x 4B
  const uint32_t lds0  = (uint32_t)(uintptr_t)(&tile[PAD]); // LDS byte offset
  const uint64_t sbase = (uint64_t)(uintptr_t)(S + rbase);  // uniform base
#pragma unroll
  for (int it = 0; it < T_DIM / 256; ++it) {
    uint32_t col   = tid + it * 256u;
    uint32_t laddr = lds0 + col * 4u;   // LDS destination byte address
    uint32_t moff  = col * 4u;          // per-lane global byte offset (GVS)
    asm volatile("global_load_async_to_lds_b32 %0, %1, %2"
                 :: "v"(laddr), "v"(moff), "s"(sbase)
                 : "memory");
  }
#if __has_builtin(__builtin_amdgcn_s_wait_asynccnt)
  __builtin_amdgcn_s_wait_asynccnt(0);
#else
  asm volatile("s_wait_asynccnt 0" ::: "memory");
#endif
  __syncthreads();

#pragma unroll
  for (int j = 0; j < T_DIM / 256; ++j) {
    const int c = (int)tid + j * 256;
    float med = median31([&](int i) -> float { return tile[c + i]; });
    harm[rbase + (size_t)c] = med;
  }
}

// ---------------------------------------------------------------------------
// Kernel 2: median along F (stride T_DIM). One output per thread; lanes are
// consecutive in T so every strided load is fully coalesced; S (16.8 MB)
// lives in the 192 MB L2, so the 31x re-read is L2-bandwidth cheap.
// ---------------------------------------------------------------------------
__global__ __launch_bounds__(256) void hpss_med_freq(const float* __restrict__ S,
                                                     float* __restrict__ perc) {
  const unsigned gid = blockIdx.x * 256u + threadIdx.x;
  const unsigned t   = gid & (T_DIM - 1);
  const unsigned f   = (gid / T_DIM) % F_DIM;
  const unsigned bc  = gid / (unsigned)(T_DIM * F_DIM);
  const float* base  = S + (size_t)bc * F_DIM * T_DIM + t;

  float med = median31([&](int i) -> float {
    int fi = (int)f + i - PAD;
    return (fi >= 0 && fi < F_DIM) ? base[(size_t)fi * T_DIM] : 0.0f;
  });
  perc[gid] = med;
}

// ---------------------------------------------------------------------------
// Kernel 3: softmask + multiply, in place over the two median planes.
// mask_h = (h/Z)^2 / ((h/Z)^2 + (p/Z)^2), Z = max(h,p) (1.0 if < TINY);
// mask_p uses the identical Z and the commuted (bitwise-identical) sum.
// ---------------------------------------------------------------------------
__global__ __launch_bounds__(256) void hpss_mask(const float* __restrict__ S,
                                                 float* __restrict__ out) {
  const unsigned i = blockIdx.x * 256u + threadIdx.x;
  const float s = S[i];
  const float h = out[i];
  const float p = out[N_ELEM + i];
  float Z = fmaxf(h, p);
  Z = (Z < TINY_F) ? 1.0f : Z;
  float m = h / Z; m = m * m;
  float r = p / Z; r = r * r;
  const float den = m + r;
  out[i]          = s * (m / den);
  out[N_ELEM + i] = s * (r / den);
}

// ---------------------------------------------------------------------------
// Launch: harm medians -> out[0..N), perc medians -> out[N..2N), then mask
// overwrites both planes elementwise (read-then-write per element => safe).
// ---------------------------------------------------------------------------
extern "C" void kernel_launch(void* const* d_in, const int* in_sizes, int n_in,
                              void* d_out, int out_size, void* d_ws, size_t ws_size,
                              hipStream_t stream) {
  (void)in_sizes; (void)n_in; (void)d_ws; (void)ws_size; (void)out_size;
  const float* S = (const float*)d_in[0];
  float* out = (float*)d_out;

  const int rows = B_DIM * C_DIM * F_DIM;        // 2050
  const int nblk = N_ELEM / 256;                 // 16400 (exact)

  hpss_med_time<<<rows, 256, 0, stream>>>(S, out);            // harmonic plane
  hpss_med_freq<<<nblk, 256, 0, stream>>>(S, out + N_ELEM);   // percussive plane
  hpss_mask    <<<nblk, 256, 0, stream>>>(S, out);            // softmask both
}